// GaussianRasterizer_67525475828242
// MI455X (gfx1250) — compile-verified
//
#include <hip/hip_runtime.h>
#include <math.h>

#define N_G      1024
#define IMG_H    256
#define IMG_W    256
#define TILE     16
#define GSTRIDE  16            // floats per packed gaussian record (64 B)
#define A_MAX    0.99f
#define A_MIN    (1.0f/255.0f)

typedef float v2f __attribute__((ext_vector_type(2)));
typedef float v8f __attribute__((ext_vector_type(8)));

// ---------------------------------------------------------------------------
// Pass 1: per-gaussian conic -> polynomial coefficients of the exponent
//   power(px,py) = K0 + K1*px + K2*py + K3*px^2 + K4*px*py + K5*py^2
// plus opacity/color, conservative cull radius, and the int32 radii output.
// Record layout (floats): [K0 K1 K2 K3 | op r g b | K4 K5 mx my | cullR 0 0 0]
// ---------------------------------------------------------------------------
__global__ void gs_preprocess(const float* __restrict__ means2D,
                              const float* __restrict__ opac,
                              const float* __restrict__ colors,
                              const float* __restrict__ scales,
                              const float* __restrict__ rots,
                              float* __restrict__ gbuf,
                              int*   __restrict__ radii_out)
{
    int n = blockIdx.x * blockDim.x + threadIdx.x;
    if (n >= N_G) return;

    float mx = means2D[2*n], my = means2D[2*n+1];
    float op = opac[n];
    float cr = colors[3*n], cg = colors[3*n+1], cbl = colors[3*n+2];
    float sx = scales[2*n], sy = scales[2*n+1];
    float th = rots[n];
    float c = cosf(th), s = sinf(th);
    float sx2 = sx*sx, sy2 = sy*sy;

    float a = c*c*sx2 + s*s*sy2 + 0.3f;
    float b = c*s*(sx2 - sy2);
    float d = s*s*sx2 + c*c*sy2 + 0.3f;
    float inv = 1.0f / (a*d - b*b);
    float ca =  d*inv, cb = -b*inv, cc = a*inv;   // conic = Sigma^{-1}

    float mid = 0.5f*(a + d);
    float lam = mid + sqrtf(fmaxf(0.25f*(a-d)*(a-d) + b*b, 1e-8f)); // lam_max(Sigma)
    radii_out[n] = (int)ceilf(3.0f * sqrtf(lam));

    // alpha = op*exp(power) < 1/255 guaranteed when dist^2 > 2*ln(255*op)*lam
    float cullR = 0.0f;
    if (op > A_MIN) cullR = sqrtf(2.0f * logf(255.0f*op) * lam) + 1.0f; // +1px margin

    float K3 = -0.5f*ca, K4 = -cb, K5 = -0.5f*cc;
    float K1 = ca*mx + cb*my;
    float K2 = cb*mx + cc*my;
    float K0 = -(0.5f*ca*mx*mx + cb*mx*my + 0.5f*cc*my*my);

    float* g = gbuf + n*GSTRIDE;
    g[0]=K0;  g[1]=K1;  g[2]=K2;  g[3]=K3;
    g[4]=op;  g[5]=cr;  g[6]=cg;  g[7]=cbl;
    g[8]=K4;  g[9]=K5;  g[10]=mx; g[11]=my;
    g[12]=cullR; g[13]=0.0f; g[14]=0.0f; g[15]=0.0f;
}

// ---------------------------------------------------------------------------
// Pass 2: tile rasterizer. One 16x16 tile per block, 16 waves (one pixel-row
// per wave32). Whole gaussian table (64 KB) staged to LDS via async loads.
// Per 16-gaussian chunk: power tile = 2x V_WMMA_F32_16X16X4_F32; lane p holds
// pixel p's powers for gaussians 0..7, lane p+16 for 8..15; transmittance
// chain stitched across halves with one shfl_xor(16).
// ---------------------------------------------------------------------------
__global__ __launch_bounds__(512)
void gs_raster(const float* __restrict__ gbuf,
               const float* __restrict__ bg,
               float* __restrict__ out)
{
    __shared__ __align__(16) float lds[N_G * GSTRIDE];   // 64 KB

    const int lane = threadIdx.x;        // 0..31
    const int row  = threadIdx.y;        // 0..15
    const int tid  = row * 32 + lane;    // 0..511
    const int tX   = blockIdx.x, tY = blockIdx.y;

    // ---- stage entire gaussian table to LDS (CDNA5 async global->LDS) ----
    {
        unsigned ldsBase = (unsigned)(size_t)(&lds[0]);
        #pragma unroll
        for (int i = 0; i < 8; ++i) {
            unsigned off = (unsigned)(tid + i*512) * 16u;   // 512 thr * 8 * 16B = 64KB
            unsigned la  = ldsBase + off;
            asm volatile("global_load_async_to_lds_b128 %0, %1, %2"
                         :: "v"(la), "v"(off), "s"(gbuf) : "memory");
        }
        asm volatile("s_wait_asynccnt 0x0" ::: "memory");
    }
    __syncthreads();

    const int   hi   = lane >> 4;        // 0: gaussians 0..7 of chunk, 1: 8..15
    const int   pcol = lane & 15;        // pixel column owned by this lane
    const float px   = (float)(tX*TILE + pcol) + 0.5f;
    const float py   = (float)(tY*TILE + row)  + 0.5f;
    const float tcx  = (float)(tX*TILE) + 8.0f;
    const float tcy  = (float)(tY*TILE) + 8.0f;

    // B matrix (4xK-slice x 16 pixels), constant for the whole loop.
    // VGPR0 = rows K=0 (lo lanes) / K=2 (hi lanes); VGPR1 = K=1 / K=3.
    v2f b01, b45;
    b01[0] = hi ? py      : 1.0f;     // k=0 -> 1, k=2 -> py
    b01[1] = hi ? px*px   : px;       // k=1 -> px, k=3 -> px^2
    b45[0] = hi ? 0.0f    : px*py;    // k=4 -> px*py, k=6 -> pad
    b45[1] = hi ? 0.0f    : py*py;    // k=5 -> py^2,  k=7 -> pad

    float T = 1.0f, accR = 0.0f, accG = 0.0f, accB = 0.0f;

    for (int base = 0; base < N_G; base += 16) {
        const float* gl = &lds[(base + (lane & 15)) * GSTRIDE];

        // conservative tile cull (exact-safe: beyond cullR, alpha < 1/255 -> 0)
        float gmx = gl[10], gmy = gl[11], crad = gl[12];
        float ddx = fmaxf(fabsf(gmx - tcx) - 8.0f, 0.0f);
        float ddy = fmaxf(fabsf(gmy - tcy) - 8.0f, 0.0f);
        bool  hit = (ddx*ddx + ddy*ddy) <= crad*crad;
        if (__ballot(hit) == 0ull) continue;   // wave-uniform; skipped chunk has T unchanged

        // A matrix: rows M = 16 gaussians of the chunk, cols K = poly features
        v2f a01 = *(const v2f*)(gl + (hi ? 2 : 0));    // lo:(K0,K1) hi:(K2,K3)
        v2f a45;
        if (hi) { a45[0] = 0.0f; a45[1] = 0.0f; }       // pad rows k=6,7
        else    { a45 = *(const v2f*)(gl + 8); }        // (K4,K5)

        v8f cz = {};
        v8f dmat = __builtin_amdgcn_wmma_f32_16x16x4_f32(
                       false, a01, false, b01, (short)0, cz, false, false);
        dmat = __builtin_amdgcn_wmma_f32_16x16x4_f32(
                       false, a45, false, b45, (short)0, dmat, false, false);

        // sequential composite of this lane-half's 8 gaussians
        float prodLoc = 1.0f, lr = 0.0f, lg = 0.0f, lb = 0.0f;
        #pragma unroll
        for (int j = 0; j < 8; ++j) {
            const float4 oc = *(const float4*)(&lds[(base + hi*8 + j)*GSTRIDE + 4]);
            float pw = fminf(dmat[j], 0.0f);
            float al = fminf(oc.x * __expf(pw), A_MAX);
            al = (al < A_MIN) ? 0.0f : al;
            float w = prodLoc * al;
            lr += w * oc.y;  lg += w * oc.z;  lb += w * oc.w;
            prodLoc *= (1.0f - al);
        }
        // stitch halves: hi half's prefix transmittance includes lo's product
        float other = __shfl_xor(prodLoc, 16, 32);
        float sc    = T * (hi ? other : 1.0f);
        accR += sc * lr;  accG += sc * lg;  accB += sc * lb;
        T *= prodLoc * other;                 // identical in both halves
    }

    // combine the two lane-halves' color partials
    accR += __shfl_xor(accR, 16, 32);
    accG += __shfl_xor(accG, 16, 32);
    accB += __shfl_xor(accB, 16, 32);

    if (hi == 0) {
        int pidx = (tY*TILE + row) * IMG_W + (tX*TILE + pcol);
        out[0*IMG_H*IMG_W + pidx] = accR + bg[0]*T;
        out[1*IMG_H*IMG_W + pidx] = accG + bg[1]*T;
        out[2*IMG_H*IMG_W + pidx] = accB + bg[2]*T;
    }
}

// ---------------------------------------------------------------------------
extern "C" void kernel_launch(void* const* d_in, const int* in_sizes, int n_in,
                              void* d_out, int out_size, void* d_ws, size_t ws_size,
                              hipStream_t stream)
{
    (void)in_sizes; (void)n_in; (void)out_size; (void)ws_size;

    const float* means  = (const float*)d_in[0];
    const float* opac   = (const float*)d_in[1];
    const float* colors = (const float*)d_in[2];
    const float* scales = (const float*)d_in[3];
    const float* rots   = (const float*)d_in[4];
    const float* bg     = (const float*)d_in[5];

    float* out   = (float*)d_out;
    int*   radii = (int*)d_out + 3*IMG_H*IMG_W;   // tuple output #2 (int32 bits)
    float* gbuf  = (float*)d_ws;                  // 1024 * 64 B packed records

    gs_preprocess<<<dim3((N_G + 255)/256), dim3(256), 0, stream>>>(
        means, opac, colors, scales, rots, gbuf, radii);

    gs_raster<<<dim3(IMG_W/TILE, IMG_H/TILE), dim3(32, 16), 0, stream>>>(
        gbuf, bg, out);
}